// Model_884763263639
// MI455X (gfx1250) — compile-verified
//
#include <hip/hip_runtime.h>
#include <hip/hip_bf16.h>

#define DIN 32
#define H   16

typedef __attribute__((ext_vector_type(2))) float v2f;
typedef __attribute__((ext_vector_type(8))) float v8f;
typedef long long i64;  // edge_index is jnp.int64 in the reference

__device__ __forceinline__ float silu(float v) {
    return v / (1.0f + __expf(-v));
}

// ---------------------------------------------------------------------------
// Zero-fill (agg buffers must be re-zeroed every call: harness replays graph)
// ---------------------------------------------------------------------------
__global__ void zero_f32(float* __restrict__ p, int n) {
    int i = blockIdx.x * blockDim.x + threadIdx.x;
    if (i < n) p[i] = 0.0f;
}

// ---------------------------------------------------------------------------
// Scatter-add: agg[dst[e], :] += h[src[e], :].  One thread = (edge, 4 feats).
// Feature tables are L2-resident (<=12.8MB vs 192MB L2), so the random
// gathers + fp32 atomics stay on-chip.
// ---------------------------------------------------------------------------
template<int D>
__global__ void scatter_add(const float* __restrict__ h,
                            const i64* __restrict__ src,
                            const i64* __restrict__ dst,
                            float* __restrict__ agg, int E) {
    constexpr int G = D / 4;
    long long tid = (long long)blockIdx.x * blockDim.x + threadIdx.x;
    if (tid >= (long long)E * G) return;
    int e = (int)(tid / G);
    int g = (int)(tid % G);
    long long s = src[e];
    long long d = dst[e];
    const float4 v = *(const float4*)(h + s * (long long)D + g * 4);
    float* out = agg + d * (long long)D + g * 4;
    atomicAdd(out + 0, v.x);
    atomicAdd(out + 1, v.y);
    atomicAdd(out + 2, v.z);
    atomicAdd(out + 3, v.w);
}

// ---------------------------------------------------------------------------
// One wave computes a 16-node tile:
//   hout[16,16] = silu( agg[16,K] @ Wl^T + bl + hin[16,K] @ Wr^T )
// using V_WMMA_F32_16X16X4_F32, accumulating K in chunks of 4 (fp32-exact).
// A frag: lane m=lane&15 holds rows, vgpr pair = K {kb,kb+1}, kb = k0 + hi*2.
// B frag: lane n holds cols, same K striping. C/D: 8 vgprs, rows r / r+8.
// ---------------------------------------------------------------------------
template<int K>
__global__ void sage_layer(const float* __restrict__ agg,   // [N, K]
                           const float* __restrict__ hin,   // [N, K]
                           const float* __restrict__ Wl,    // [16, K]
                           const float* __restrict__ bl,    // [16]
                           const float* __restrict__ Wr,    // [16, K]
                           float* __restrict__ hout,        // [N, 16]
                           int N, int ntiles) {
    int wid = (blockIdx.x * blockDim.x + threadIdx.x) >> 5;   // wave id (uniform)
    if (wid >= ntiles) return;                                 // wave-uniform exit
    int lane = threadIdx.x & 31;
    int m    = lane & 15;
    int hi   = lane >> 4;
    int n    = m;
    int row  = wid * 16 + m;
    int rowc = row < N ? row : N - 1;   // clamp (keeps EXEC full for WMMA)

    v8f c = {};
#pragma unroll
    for (int k0 = 0; k0 < K; k0 += 4) {
        int kb = k0 + hi * 2;
        v2f a = *(const v2f*)(agg + (size_t)rowc * K + kb);
        v2f b; b.x = Wl[n * K + kb]; b.y = Wl[n * K + kb + 1];
        c = __builtin_amdgcn_wmma_f32_16x16x4_f32(false, a, false, b,
                                                  (short)0, c, false, false);
    }
#pragma unroll
    for (int k0 = 0; k0 < K; k0 += 4) {
        int kb = k0 + hi * 2;
        v2f a = *(const v2f*)(hin + (size_t)rowc * K + kb);
        v2f b; b.x = Wr[n * K + kb]; b.y = Wr[n * K + kb + 1];
        c = __builtin_amdgcn_wmma_f32_16x16x4_f32(false, a, false, b,
                                                  (short)0, c, false, false);
    }
    float bias = bl[n];
#pragma unroll
    for (int r = 0; r < 8; ++r) {
        int orow = wid * 16 + hi * 8 + r;
        if (orow < N) hout[(size_t)orow * H + n] = silu(c[r] + bias);
    }
}

// ---------------------------------------------------------------------------
// Dense layer [N,K] -> [N,32] with SiLU: two 16-wide output halves per wave.
// ---------------------------------------------------------------------------
template<int K>
__global__ void dense32_silu(const float* __restrict__ in,  // [N, K]
                             const float* __restrict__ W,   // [32, K]
                             const float* __restrict__ b,   // [32]
                             float* __restrict__ out,       // [N, 32]
                             int N, int ntiles) {
    int wid = (blockIdx.x * blockDim.x + threadIdx.x) >> 5;
    if (wid >= ntiles) return;
    int lane = threadIdx.x & 31;
    int m    = lane & 15;
    int hi   = lane >> 4;
    int n    = m;
    int row  = wid * 16 + m;
    int rowc = row < N ? row : N - 1;

    v8f c0 = {}, c1 = {};
#pragma unroll
    for (int k0 = 0; k0 < K; k0 += 4) {
        int kb = k0 + hi * 2;
        v2f a = *(const v2f*)(in + (size_t)rowc * K + kb);
        v2f b0; b0.x = W[n * K + kb];        b0.y = W[n * K + kb + 1];
        v2f b1; b1.x = W[(16 + n) * K + kb]; b1.y = W[(16 + n) * K + kb + 1];
        c0 = __builtin_amdgcn_wmma_f32_16x16x4_f32(false, a, false, b0,
                                                   (short)0, c0, false, false);
        c1 = __builtin_amdgcn_wmma_f32_16x16x4_f32(false, a, false, b1,
                                                   (short)0, c1, false, false);
    }
    float bias0 = b[n], bias1 = b[16 + n];
#pragma unroll
    for (int r = 0; r < 8; ++r) {
        int orow = wid * 16 + hi * 8 + r;
        if (orow < N) {
            out[(size_t)orow * 32 + n]      = silu(c0[r] + bias0);
            out[(size_t)orow * 32 + 16 + n] = silu(c1[r] + bias1);
        }
    }
}

// ---------------------------------------------------------------------------
// Final projection [N,32] -> [N,1] (tiny dot product, VALU is fine).
// ---------------------------------------------------------------------------
__global__ void dense1(const float* __restrict__ in,   // [N, 32]
                       const float* __restrict__ W3,   // [1, 32]
                       const float* __restrict__ b3,   // [1]
                       float* __restrict__ out, int N) {
    int i = blockIdx.x * blockDim.x + threadIdx.x;
    if (i >= N) return;
    float acc = b3[0];
#pragma unroll
    for (int k = 0; k < 32; ++k) acc += in[(size_t)i * 32 + k] * W3[k];
    out[i] = acc;
}

// ---------------------------------------------------------------------------
extern "C" void kernel_launch(void* const* d_in, const int* in_sizes, int n_in,
                              void* d_out, int out_size, void* d_ws, size_t ws_size,
                              hipStream_t stream) {
    (void)n_in; (void)out_size; (void)ws_size;

    const float* x   = (const float*)d_in[0];
    const i64*   ei  = (const i64*)  d_in[1];
    const float* Wl0 = (const float*)d_in[2];
    const float* bl0 = (const float*)d_in[3];
    const float* Wr0 = (const float*)d_in[4];
    const float* Wls = (const float*)d_in[5];
    const float* bls = (const float*)d_in[6];
    const float* Wrs = (const float*)d_in[7];
    const float* W1  = (const float*)d_in[8];
    const float* b1  = (const float*)d_in[9];
    const float* W2  = (const float*)d_in[10];
    const float* b2  = (const float*)d_in[11];
    const float* W3  = (const float*)d_in[12];
    const float* b3  = (const float*)d_in[13];

    const int N = in_sizes[0] / DIN;     // 100000
    const int E = in_sizes[1] / 2;       // 1600000
    const i64* src = ei;                 // edge_index[0]
    const i64* dst = ei + E;             // edge_index[1]

    // Workspace layout (floats), 64*N total = 25.6 MB:
    //   [0      , 32N) : agg (layer 0 uses 32N, later layers 16N) -> reused as out1
    //   [32N    , 48N) : hA   -> (with hB) reused as out2 [N,32]
    //   [48N    , 64N) : hB
    float* ws   = (float*)d_ws;
    float* agg  = ws;
    float* hA   = ws + (size_t)N * 32;
    float* hB   = hA + (size_t)N * 16;
    float* out1 = agg;   // agg dead after last sage layer
    float* out2 = hA;    // hA+hB dead after mlp1

    const int ntiles = (N + 15) / 16;
    dim3 blk(256);
    dim3 gTile((unsigned)((ntiles * 32 + 255) / 256));

    // ---- Layer 0: d = 32 --------------------------------------------------
    zero_f32<<<dim3((unsigned)((N * 32 + 255) / 256)), blk, 0, stream>>>(agg, N * 32);
    {
        long long nthr = (long long)E * 8;
        scatter_add<32><<<dim3((unsigned)((nthr + 255) / 256)), blk, 0, stream>>>(
            x, src, dst, agg, E);
    }
    sage_layer<32><<<gTile, blk, 0, stream>>>(agg, x, Wl0, bl0, Wr0, hA, N, ntiles);

    // ---- Layers 1..3: d = 16 ---------------------------------------------
    for (int i = 0; i < 3; ++i) {
        const float* hin = (i % 2 == 0) ? hA : hB;
        float*       hot = (i % 2 == 0) ? hB : hA;
        zero_f32<<<dim3((unsigned)((N * 16 + 255) / 256)), blk, 0, stream>>>(agg, N * 16);
        long long nthr = (long long)E * 4;
        scatter_add<16><<<dim3((unsigned)((nthr + 255) / 256)), blk, 0, stream>>>(
            hin, src, dst, agg, E);
        sage_layer<16><<<gTile, blk, 0, stream>>>(agg, hin,
                                                  Wls + i * H * H, bls + i * H,
                                                  Wrs + i * H * H, hot, N, ntiles);
    }
    // final hidden state is in hB (layers 1,2,3 end hA->hB->hA->hB)

    // ---- Output MLP -------------------------------------------------------
    dense32_silu<16><<<gTile, blk, 0, stream>>>(hB, W1, b1, out1, N, ntiles);
    dense32_silu<32><<<gTile, blk, 0, stream>>>(out1, W2, b2, out2, N, ntiles);
    dense1<<<dim3((unsigned)((N + 255) / 256)), blk, 0, stream>>>(out2, W3, b3,
                                                                  (float*)d_out, N);
}